// UnionLossNet_76922864271915
// MI455X (gfx1250) — compile-verified
//
#include <hip/hip_runtime.h>

#define BLOCK 256
#define EPSV 1e-5f

// ---------------------------------------------------------------------------
// Workspace layout (d_ws):
//   [0..3]   : u32 min-bits of margin candidates (seeded with bits of 1.0f,
//              which folds in the min(1.0, ...) clamp)
//   [16..31] : two f64 accumulators (asso sum, attr sum)
// ---------------------------------------------------------------------------

__global__ void ul_init(unsigned int* __restrict__ minbits,
                        double* __restrict__ sums) {
  minbits[0] = 0x3F800000u;  // 1.0f
  sums[0] = 0.0;
  sums[1] = 0.0;
}

__device__ __forceinline__ float pos_min4(float4 p, float4 t, float m) {
  if (fabsf(t.x - 1.0f) < EPSV) m = fminf(m, p.x);
  if (fabsf(t.y - 1.0f) < EPSV) m = fminf(m, p.y);
  if (fabsf(t.z - 1.0f) < EPSV) m = fminf(m, p.z);
  if (fabsf(t.w - 1.0f) < EPSV) m = fminf(m, p.w);
  return m;
}

// Pass 1: margin = min(1, min(predict where target==1)) over BOTH tensors.
__global__ __launch_bounds__(BLOCK) void ul_margin(
    const float4* __restrict__ ap, const float4* __restrict__ at,
    const float4* __restrict__ tp, const float4* __restrict__ tt,
    unsigned int* __restrict__ minbits, int nvec) {
  const int stride = gridDim.x * BLOCK;
  float m = 1.0f;
  for (int i = blockIdx.x * BLOCK + threadIdx.x; i < nvec; i += stride) {
    if (i + stride < nvec) {  // deep prefetch of next grid-stride slab into L2
      __builtin_prefetch(&ap[i + stride], 0, 3);
      __builtin_prefetch(&at[i + stride], 0, 3);
      __builtin_prefetch(&tp[i + stride], 0, 3);
      __builtin_prefetch(&tt[i + stride], 0, 3);
    }
    m = pos_min4(ap[i], at[i], m);
    m = pos_min4(tp[i], tt[i], m);
  }
  // wave32 min-reduce (CDNA5: 5 xor steps)
#pragma unroll
  for (int off = 16; off > 0; off >>= 1) m = fminf(m, __shfl_xor(m, off, 32));
  __shared__ float smin[BLOCK / 32];
  const int lane = threadIdx.x & 31;
  const int wv = threadIdx.x >> 5;
  if (lane == 0) smin[wv] = m;
  __syncthreads();
  if (threadIdx.x == 0) {
    float bm = smin[0];
#pragma unroll
    for (int w = 1; w < BLOCK / 32; ++w) bm = fminf(bm, smin[w]);
    // All candidates are >= 0, so u32 bit-order == float order.
    atomicMin(minbits, __float_as_uint(bm));
  }
}

__device__ __forceinline__ float sl1(float p, float t, float m) {
  const bool is_pos = fabsf(t - 1.0f) < EPSV;
  if (is_pos || p < m) return 0.0f;  // zero_mask branch
  const float d = p - m;             // >= 0 here, so |p*1 - m| == p - m
  return (d < 1.0f) ? 0.5f * d * d : d - 0.5f;  // BETA = 1
}

__device__ __forceinline__ float sl1v(float4 p, float4 t, float m) {
  return sl1(p.x, t.x, m) + sl1(p.y, t.y, m) + sl1(p.z, t.z, m) +
         sl1(p.w, t.w, m);
}

// Pass 2: smooth-L1 sums for both tensors.
__global__ __launch_bounds__(BLOCK) void ul_loss(
    const float4* __restrict__ ap, const float4* __restrict__ at,
    const float4* __restrict__ tp, const float4* __restrict__ tt,
    const unsigned int* __restrict__ minbits, double* __restrict__ sums,
    int nvec) {
  const float margin = __uint_as_float(*minbits);  // uniform scalar load
  const int stride = gridDim.x * BLOCK;
  float s0 = 0.0f, s1 = 0.0f;
  for (int i = blockIdx.x * BLOCK + threadIdx.x; i < nvec; i += stride) {
    if (i + stride < nvec) {
      __builtin_prefetch(&ap[i + stride], 0, 3);
      __builtin_prefetch(&at[i + stride], 0, 3);
      __builtin_prefetch(&tp[i + stride], 0, 3);
      __builtin_prefetch(&tt[i + stride], 0, 3);
    }
    s0 += sl1v(ap[i], at[i], margin);
    s1 += sl1v(tp[i], tt[i], margin);
  }
#pragma unroll
  for (int off = 16; off > 0; off >>= 1) {
    s0 += __shfl_xor(s0, off, 32);
    s1 += __shfl_xor(s1, off, 32);
  }
  __shared__ float ss0[BLOCK / 32];
  __shared__ float ss1[BLOCK / 32];
  const int lane = threadIdx.x & 31;
  const int wv = threadIdx.x >> 5;
  if (lane == 0) {
    ss0[wv] = s0;
    ss1[wv] = s1;
  }
  __syncthreads();
  if (threadIdx.x == 0) {
    float b0 = ss0[0], b1 = ss1[0];
#pragma unroll
    for (int w = 1; w < BLOCK / 32; ++w) {
      b0 += ss0[w];
      b1 += ss1[w];
    }
    atomicAdd(&sums[0], (double)b0);  // global_atomic_add_f64
    atomicAdd(&sums[1], (double)b1);
  }
}

__global__ void ul_final(const double* __restrict__ sums,
                         float* __restrict__ out, double inv_n) {
  out[0] = (float)(sums[0] * inv_n);
  out[1] = (float)(sums[1] * inv_n);
}

extern "C" void kernel_launch(void* const* d_in, const int* in_sizes, int n_in,
                              void* d_out, int out_size, void* d_ws,
                              size_t ws_size, hipStream_t stream) {
  const float4* ap = (const float4*)d_in[0];  // asso_predict
  const float4* at = (const float4*)d_in[1];  // asso_target
  const float4* tp = (const float4*)d_in[2];  // attr_predict
  const float4* tt = (const float4*)d_in[3];  // attr_target
  const int n = in_sizes[0];  // 131072*128 = 16,777,216 (divisible by 4)
  const int nvec = n / 4;

  unsigned int* minbits = (unsigned int*)d_ws;
  double* sums = (double*)((char*)d_ws + 16);
  float* out = (float*)d_out;

  int grid = (nvec + BLOCK - 1) / BLOCK;
  if (grid > 4096) grid = 4096;

  hipLaunchKernelGGL(ul_init, dim3(1), dim3(1), 0, stream, minbits, sums);
  hipLaunchKernelGGL(ul_margin, dim3(grid), dim3(BLOCK), 0, stream, ap, at, tp,
                     tt, minbits, nvec);
  hipLaunchKernelGGL(ul_loss, dim3(grid), dim3(BLOCK), 0, stream, ap, at, tp,
                     tt, minbits, sums, nvec);
  hipLaunchKernelGGL(ul_final, dim3(1), dim3(1), 0, stream, sums, out,
                     1.0 / (double)n);
}